// SeqCrossEntropyLoss_24842090840646
// MI455X (gfx1250) — compile-verified
//
#include <hip/hip_runtime.h>
#include <hip/hip_bf16.h>

// SeqCrossEntropyLoss for B=2048, T=1024, V=33, EOS=0.
// main_len = min(eos+1, T); tail (if n-1 > eos) needs only row eos + labels.
// eos ~ Geom(1/33) => expected pred traffic ~35 rows/batch, not 1024.
// Chunk staging now uses CDNA5 GLOBAL_LOAD_ASYNC_TO_LDS_B128 (ASYNCcnt-tracked,
// LDS written directly from the memory hierarchy, no VGPR round-trip).

#define B_SIZE 2048
#define T_MAX  1024
#define V_DIM  33
#define BT     128      // threads per block = 4 wave32
#define CHUNK  128      // pred rows staged per iteration
#define SENT   0x7fffffff

typedef float v2f __attribute__((ext_vector_type(2)));
typedef float v8f __attribute__((ext_vector_type(8)));

// Sum 32 per-lane f32 values across a wave with one V_WMMA_F32_16X16X4_F32.
// A: lane L<16 -> A[L][0]=s; lane L>=16 -> A[L-16][2]=s (other K slots 0).
// B = all-ones 4x16  =>  D[m][n] = s_m + s_{m+16} for every n.
// Lane L<16 holds D[0..7][L] in 8 vgprs; lane L>=16 holds D[8..15][L-16].
// (sum of 8 vgprs @ lane0) + (same @ lane16) = full wave sum. Exact f32 adds.
// Requires full EXEC (call from a uniformly-active wave).
__device__ __forceinline__ float wave_reduce_wmma(float s) {
    v2f a;    a.x = s;    a.y = 0.0f;
    v2f ones; ones.x = 1.0f; ones.y = 1.0f;
    v8f c = {};
    v8f d = __builtin_amdgcn_wmma_f32_16x16x4_f32(
        /*neg_a=*/false, a, /*neg_b=*/false, ones,
        /*c_mod=*/(short)0, c, /*reuse_a=*/false, /*reuse_b=*/false);
    float r = d[0] + d[1] + d[2] + d[3] + d[4] + d[5] + d[6] + d[7];
    return __shfl(r, 0, 32) + __shfl(r, 16, 32);
}

__global__ __launch_bounds__(BT) void seq_ce_per_batch(
    const float* __restrict__ preds, const int* __restrict__ labels,
    const int* __restrict__ sizes, float* __restrict__ partial)
{
    __shared__ __align__(16) float rowbuf[CHUNK * V_DIM];   // 16896 B stage
    __shared__ float xe[V_DIM];               // logits of the eos row
    __shared__ float s_lse_e;
    __shared__ int   s_e;
    __shared__ float red[BT];

    const int b   = blockIdx.x;
    const int tid = threadIdx.x;
    const int n   = sizes[b];
    const float* P  = preds  + (size_t)b * T_MAX * V_DIM;
    const int*   Lb = labels + (size_t)b * T_MAX;

    float acc = 0.0f;   // this thread's sum of (lse - x[tgt]) over main rows
    int   e = T_MAX;
    bool  found = false;

    for (int cix = 0; cix < T_MAX / CHUNK; ++cix) {
        const int t0 = cix * CHUNK;
        if (tid == 0) s_e = SENT;

        // Async stage: CHUNK*V_DIM = 4224 floats = 1056 b128 transfers,
        // coalesced across the wave. Chunk byte offset is a multiple of
        // 128*132 = 16896 -> 16B aligned on both sides.
        {
            const float4* src4 = (const float4*)(P + (size_t)t0 * V_DIM);
            const float4* dst4 = (const float4*)rowbuf;
            for (int i = tid; i < (CHUNK * V_DIM) / 4; i += BT) {
                // Generic LDS-aperture addr: low 32 bits == LDS byte offset.
                uint32_t lds_off = (uint32_t)(uintptr_t)(dst4 + i);
                uint64_t gaddr   = (uint64_t)(uintptr_t)(src4 + i);
                asm volatile("global_load_async_to_lds_b128 %0, %1, off"
                             :: "v"(lds_off), "v"(gaddr) : "memory");
            }
            // Each wave drains its own ASYNCcnt before signaling the barrier,
            // so post-barrier every wave sees all LDS writes.
            asm volatile("s_wait_asynccnt 0" ::: "memory");
        }
        __syncthreads();

        const int t = t0 + tid;                 // one row per thread
        const float* x = rowbuf + tid * V_DIM;  // stride 33: bank-conflict-free
        float m = x[0];
        #pragma unroll
        for (int v = 1; v < V_DIM; ++v) m = fmaxf(m, x[v]);
        const bool iseos = (x[0] == m);         // argmax==0 (first-index tie-break)
        float sexp = 0.0f;
        #pragma unroll
        for (int v = 0; v < V_DIM; ++v) sexp += __expf(x[v] - m);
        const float lse = m + __logf(sexp);
        const int  tgt = Lb[min(t, n - 1)];
        const float ll = lse - x[tgt];          // positive CE term for row t

        if (iseos) atomicMin(&s_e, t);          // int atomic: order-independent
        __syncthreads();
        const int ce = s_e;
        if (t <= ce) acc += ll;                 // rows 0..eos contribute (SENT -> all)
        if (t == ce) {                          // cache eos row for the tail
            s_lse_e = lse;
            for (int v = 0; v < V_DIM; ++v) xe[v] = x[v];
        }
        if (ce != SENT) { e = ce; found = true; break; }  // uniform exit
        __syncthreads();                        // rowbuf reuse guard
    }
    __syncthreads();                            // publish xe / s_lse_e

    // tail: t in [e+1, n-1] reuses row e with label[t]
    if (found && (n - 1) > e) {
        float tx = 0.0f;
        for (int t = e + 1 + tid; t <= n - 1; t += BT) tx += xe[Lb[t]];
        acc -= tx;
        if (tid == 0) acc += (float)(n - 1 - e) * s_lse_e;
    }

    red[tid] = acc;
    __syncthreads();
    if (tid < 32) {                             // wave 0, full EXEC
        float s = red[tid] + red[tid + 32] + red[tid + 64] + red[tid + 96];
        float tot = wave_reduce_wmma(s);
        if (tid == 0) partial[b] = tot;
    }
}

__global__ __launch_bounds__(256) void seq_ce_final(
    const float* __restrict__ partial, float* __restrict__ out)
{
    __shared__ float red[256];
    const int tid = threadIdx.x;
    float s = 0.0f;
    for (int i = tid; i < B_SIZE; i += 256) s += partial[i];
    red[tid] = s;
    __syncthreads();
    if (tid < 32) {
        float v = 0.0f;
        #pragma unroll
        for (int k = 0; k < 8; ++k) v += red[tid + 32 * k];
        float tot = wave_reduce_wmma(v);
        if (tid == 0) out[0] = tot;   // acc already holds the summed CE
    }
}

extern "C" void kernel_launch(void* const* d_in, const int* in_sizes, int n_in,
                              void* d_out, int out_size, void* d_ws, size_t ws_size,
                              hipStream_t stream) {
    (void)in_sizes; (void)n_in; (void)out_size; (void)ws_size;
    const float* preds  = (const float*)d_in[0];
    const int*   labels = (const int*)d_in[1];
    const int*   sizes  = (const int*)d_in[2];
    float* partial = (float*)d_ws;            // B_SIZE floats of scratch

    seq_ce_per_batch<<<B_SIZE, BT, 0, stream>>>(preds, labels, sizes, partial);
    seq_ce_final<<<1, 256, 0, stream>>>(partial, (float*)d_out);
}